// PointCloudAttentionLayer_77481210020635
// MI455X (gfx1250) — compile-verified
//
#include <hip/hip_runtime.h>
#include <math.h>

#define NNODES 65536
#define HDIM   128
#define NHEADS 8
#define DHEAD  16
#define NEDGE  1048576
#define HIDDEN 256

typedef __attribute__((ext_vector_type(2))) float v2f;
typedef __attribute__((ext_vector_type(8))) float v8f;

// ---------------- WMMA helper: D = A(16x4) * B(4x16) + C, fp32 ----------------
__device__ __forceinline__ v8f wmma4(v2f a, v2f b, v8f c) {
  return __builtin_amdgcn_wmma_f32_16x16x4_f32(
      /*neg_a=*/false, a, /*neg_b=*/false, b,
      /*c_mod=*/(short)0, c, /*reuse_a=*/false, /*reuse_b=*/false);
}

__device__ __forceinline__ float gelu_tanh(float v) {
  float v3 = v * v * v;
  return 0.5f * v * (1.0f + tanhf(0.7978845608028654f * (v + 0.044715f * v3)));
}

// order-preserving float<->uint encoding for atomic segment-max
__device__ __forceinline__ unsigned fenc(float f) {
  unsigned u = __float_as_uint(f);
  return (u & 0x80000000u) ? ~u : (u | 0x80000000u);
}
__device__ __forceinline__ float fdec(unsigned u) {
  unsigned b = (u & 0x80000000u) ? (u ^ 0x80000000u) : ~u;
  return __uint_as_float(b);
}

// LDS byte offset of a shared-memory pointer: flat shared-aperture addresses
// carry the 0-based LDS offset in their low 32 bits (CDNA5 aperture mapping).
__device__ __forceinline__ unsigned lds_off32(const void* p) {
  return (unsigned)(uintptr_t)p;
}

// Async memory->LDS copy of one 16-byte chunk (tracked by ASYNCcnt).
__device__ __forceinline__ void async_cp16(unsigned ldsByteOff, const float* gaddr) {
  asm volatile("global_load_async_to_lds_b128 %0, %1, off"
               :: "v"(ldsByteOff), "v"(gaddr)
               : "memory");
}
__device__ __forceinline__ void async_wait0() {
  asm volatile("s_wait_asynccnt 0x0" ::: "memory");
}

// ---------------- LayerNorm: one wave32 per row of 128 ----------------
__global__ __launch_bounds__(256)
void ln_kernel(const float* __restrict__ x, const float* __restrict__ g,
               const float* __restrict__ b, float* __restrict__ out) {
  int row  = (blockIdx.x * blockDim.x + threadIdx.x) >> 5;
  int lane = threadIdx.x & 31;
  if (row >= NNODES) return;
  const float4* rp = (const float4*)(x + (size_t)row * HDIM);
  float4 v = rp[lane];
  float s = v.x + v.y + v.z + v.w;
  for (int m = 16; m >= 1; m >>= 1) s += __shfl_xor(s, m, 32);
  float mu = s * (1.0f / HDIM);
  float dx = v.x - mu, dy = v.y - mu, dz = v.z - mu, dw = v.w - mu;
  float q = dx * dx + dy * dy + dz * dz + dw * dw;
  for (int m = 16; m >= 1; m >>= 1) q += __shfl_xor(q, m, 32);
  float rstd = rsqrtf(q * (1.0f / HDIM) + 1e-5f);
  float4 gg = ((const float4*)g)[lane];
  float4 bb = ((const float4*)b)[lane];
  float4 o;
  o.x = dx * rstd * gg.x + bb.x;
  o.y = dy * rstd * gg.y + bb.y;
  o.z = dz * rstd * gg.z + bb.z;
  o.w = dw * rstd * gg.w + bb.w;
  ((float4*)(out + (size_t)row * HDIM))[lane] = o;
}

// ---------------- Tiled WMMA GEMM: out[64rows x NC] = A @ W + bias (+res, act) ----
// Block: 128 threads = 4 waves; each wave computes a 16xNC strip.
// WLDS=1: stage W in LDS via async copies (square 128x128 weights only).
template <int KD, int NC, int ACT, int ADDRES, int WLDS>
__global__ __launch_bounds__(128)
void gemm_kernel(const float* __restrict__ A, const float* __restrict__ W,
                 const float* __restrict__ bias, const float* __restrict__ res,
                 float* __restrict__ out) {
  constexpr int LDSTR = KD + 4;  // pad: stride % 64 banks == 4 -> conflict free
  constexpr int WSTR  = NC + 4;
  __shared__ float ldsA[64 * LDSTR];
  __shared__ float ldsW[WLDS ? (KD * WSTR) : 4];

  const int rowBase = blockIdx.x * 64;

  // ---- async stage A panel (64 x KD) into LDS, 16B per lane per iter ----
  {
    const unsigned aBase = lds_off32(&ldsA[0]);
    constexpr int C4 = KD / 4;
    for (int idx = threadIdx.x; idx < 64 * C4; idx += 128) {
      int r  = idx / C4;
      int c  = (idx - r * C4) * 4;
      async_cp16(aBase + (unsigned)((r * LDSTR + c) * 4),
                 A + (size_t)(rowBase + r) * KD + c);
    }
    if constexpr (WLDS) {
      const unsigned wBase = lds_off32(&ldsW[0]);
      constexpr int W4 = NC / 4;
      for (int idx = threadIdx.x; idx < KD * W4; idx += 128) {
        int r = idx / W4;
        int c = (idx - r * W4) * 4;
        async_cp16(wBase + (unsigned)((r * WSTR + c) * 4),
                   W + (size_t)r * NC + c);
      }
    }
    async_wait0();
  }
  __syncthreads();

  const int wave = threadIdx.x >> 5;
  const int lane = threadIdx.x & 31;
  const int n    = lane & 15;            // A row (M) = C col (N) = lane&15
  const int kk   = (lane >> 4) << 1;     // lanes 16-31 carry K+2,K+3
  const int wrow = wave * 16;
  const int mrow = rowBase + wrow + ((lane >> 4) << 3);  // C: lanes16-31 -> M+8

  for (int ct = 0; ct < NC / 16; ++ct) {
    const int colBase = ct * 16;
    float bval = bias[colBase + n];
    v8f c = {bval, bval, bval, bval, bval, bval, bval, bval};
#pragma unroll 4
    for (int kt = 0; kt < KD / 4; ++kt) {
      const int k0 = kt * 4;
      v2f av, bv;
      av.x = ldsA[(wrow + n) * LDSTR + k0 + kk];
      av.y = ldsA[(wrow + n) * LDSTR + k0 + kk + 1];
      if constexpr (WLDS) {
        bv.x = ldsW[(k0 + kk) * WSTR + colBase + n];
        bv.y = ldsW[(k0 + kk + 1) * WSTR + colBase + n];
      } else {
        bv.x = W[(size_t)(k0 + kk) * NC + colBase + n];
        bv.y = W[(size_t)(k0 + kk + 1) * NC + colBase + n];
      }
      c = wmma4(av, bv, c);
    }
#pragma unroll
    for (int i = 0; i < 8; ++i) {
      float val = c[i];
      if constexpr (ACT == 1) val = gelu_tanh(val);
      size_t o = (size_t)(mrow + i) * NC + colBase + n;
      if constexpr (ADDRES) val += res[o];
      out[o] = val;
    }
  }
}

// ---------------- segment-softmax init ----------------
__global__ __launch_bounds__(256)
void init_md_kernel(unsigned* __restrict__ menc, float* __restrict__ denom) {
  int tid = blockIdx.x * blockDim.x + threadIdx.x;
  if (tid < NNODES * NHEADS) { menc[tid] = 0u; denom[tid] = 0.0f; }
}

__global__ __launch_bounds__(256)
void zero_kernel(float* __restrict__ p, int n) {
  int tid = blockIdx.x * blockDim.x + threadIdx.x;
  if (tid < n) p[tid] = 0.0f;
}

// ---------------- SDDMM + segment max: one thread per (edge, head) ----------------
__global__ __launch_bounds__(256)
void sddmm_kernel(const float* __restrict__ q, const float* __restrict__ k,
                  const int* __restrict__ er, const int* __restrict__ ec,
                  float* __restrict__ scores, unsigned* __restrict__ menc) {
  int tid = blockIdx.x * blockDim.x + threadIdx.x;
  int e = tid >> 3, h = tid & 7;
  if (e >= NEDGE) return;
  int r = er[e], c = ec[e];
  const float4* qp = (const float4*)(q + (size_t)r * HDIM + h * DHEAD);
  const float4* kp = (const float4*)(k + (size_t)c * HDIM + h * DHEAD);
  float s = 0.0f;
#pragma unroll
  for (int i = 0; i < 4; ++i) {
    float4 a = qp[i], b = kp[i];
    s += a.x * b.x + a.y * b.y + a.z * b.z + a.w * b.w;
  }
  s *= 0.25f;  // 1/sqrt(16)
  scores[tid] = s;
  atomicMax(&menc[r * NHEADS + h], fenc(s));
}

// ---------------- exp(s - m) and segment sum ----------------
__global__ __launch_bounds__(256)
void expsum_kernel(float* __restrict__ scores, const int* __restrict__ er,
                   const unsigned* __restrict__ menc, float* __restrict__ denom) {
  int tid = blockIdx.x * blockDim.x + threadIdx.x;
  int e = tid >> 3, h = tid & 7;
  if (e >= NEDGE) return;
  int r = er[e];
  float mval = fdec(menc[r * NHEADS + h]);
  float val = expf(scores[tid] - mval);
  scores[tid] = val;
  atomicAdd(&denom[r * NHEADS + h], val);
}

// ---------------- SpMM aggregation ----------------
__global__ __launch_bounds__(256)
void spmm_kernel(const float* __restrict__ scores, const float* __restrict__ denom,
                 const int* __restrict__ er, const int* __restrict__ ec,
                 const float* __restrict__ v, float* __restrict__ agg) {
  int tid = blockIdx.x * blockDim.x + threadIdx.x;
  int e = tid >> 3, h = tid & 7;
  if (e >= NEDGE) return;
  int r = er[e], c = ec[e];
  float alpha = scores[tid] / (denom[r * NHEADS + h] + 1e-9f);
  const float4* vp = (const float4*)(v + (size_t)c * HDIM + h * DHEAD);
  float* ap = agg + (size_t)r * HDIM + h * DHEAD;
#pragma unroll
  for (int i = 0; i < 4; ++i) {
    float4 vv = vp[i];
    atomicAdd(ap + 4 * i + 0, alpha * vv.x);
    atomicAdd(ap + 4 * i + 1, alpha * vv.y);
    atomicAdd(ap + 4 * i + 2, alpha * vv.z);
    atomicAdd(ap + 4 * i + 3, alpha * vv.w);
  }
}

extern "C" void kernel_launch(void* const* d_in, const int* in_sizes, int n_in,
                              void* d_out, int out_size, void* d_ws, size_t ws_size,
                              hipStream_t stream) {
  const float* x     = (const float*)d_in[0];
  const int*   er    = (const int*)d_in[1];
  const int*   ec    = (const int*)d_in[2];
  const float* ln1_g = (const float*)d_in[3];
  const float* ln1_b = (const float*)d_in[4];
  const float* wq = (const float*)d_in[5];
  const float* bq = (const float*)d_in[6];
  const float* wk = (const float*)d_in[7];
  const float* bk = (const float*)d_in[8];
  const float* wv = (const float*)d_in[9];
  const float* bv = (const float*)d_in[10];
  const float* wo = (const float*)d_in[11];
  const float* bo = (const float*)d_in[12];
  const float* ln2_g = (const float*)d_in[13];
  const float* ln2_b = (const float*)d_in[14];
  const float* w1 = (const float*)d_in[15];
  const float* b1 = (const float*)d_in[16];
  const float* w2 = (const float*)d_in[17];
  const float* b2 = (const float*)d_in[18];
  float* out = (float*)d_out;

  // workspace layout (floats), slots aliased by lifetime
  const size_t NH = (size_t)NNODES * HDIM;  // 8388608, == NEDGE*NHEADS
  float* ws = (float*)d_ws;
  float* xn     = ws + 0 * NH;   // later reused as agg
  float* agg    = xn;
  float* qbuf   = ws + 1 * NH;   // later reused as h
  float* hbuf   = qbuf;
  float* kbuf   = ws + 2 * NH;   // later reused as hn
  float* hnbuf  = kbuf;
  float* vbuf   = ws + 3 * NH;   // [3NH, 5NH) later reused as mlp hidden t
  float* tbuf   = vbuf;
  float* scores = ws + 4 * NH;
  unsigned* menc = (unsigned*)(ws + 5 * NH);
  float* denom   = ws + 5 * NH + (size_t)NNODES * NHEADS;

  const dim3 b256(256), b128(128);
  const int gRows  = NNODES / 64;                 // 1024 gemm blocks
  const int gLN    = NNODES / 8;                  // 8 rows per 256-thr block
  const int gEdge  = (NEDGE * NHEADS) / 256;      // 32768
  const int gMD    = (NNODES * NHEADS + 255) / 256;
  const int gAgg   = (int)(NH / 256);

  // segment state init (must run every call: ws is not re-poisoned)
  init_md_kernel<<<gMD, b256, 0, stream>>>(menc, denom);

  // 1) pre-LN
  ln_kernel<<<gLN, b256, 0, stream>>>(x, ln1_g, ln1_b, xn);

  // 2) QKV projections (fp32 WMMA, W staged in LDS via async copies)
  gemm_kernel<128, 128, 0, 0, 1><<<gRows, b128, 0, stream>>>(xn, wq, bq, nullptr, qbuf);
  gemm_kernel<128, 128, 0, 0, 1><<<gRows, b128, 0, stream>>>(xn, wk, bk, nullptr, kbuf);
  gemm_kernel<128, 128, 0, 0, 1><<<gRows, b128, 0, stream>>>(xn, wv, bv, nullptr, vbuf);

  // 3) SDDMM + segment softmax
  sddmm_kernel<<<gEdge, b256, 0, stream>>>(qbuf, kbuf, er, ec, scores, menc);
  expsum_kernel<<<gEdge, b256, 0, stream>>>(scores, er, menc, denom);

  // 4) weighted aggregation (agg aliases xn -> zero after QKV GEMMs)
  zero_kernel<<<gAgg, b256, 0, stream>>>(agg, (int)NH);
  spmm_kernel<<<gEdge, b256, 0, stream>>>(scores, denom, er, ec, vbuf, agg);

  // 5) output projection + residual: h = x + agg@wo + bo   (h aliases q)
  gemm_kernel<128, 128, 0, 1, 1><<<gRows, b128, 0, stream>>>(agg, wo, bo, x, hbuf);

  // 6) MLP block: hn = LN(h); t = gelu(hn@w1+b1); out = h + t@w2 + b2
  ln_kernel<<<gLN, b256, 0, stream>>>(hbuf, ln2_g, ln2_b, hnbuf);
  gemm_kernel<128, 256, 1, 0, 0><<<gRows, b128, 0, stream>>>(hnbuf, w1, b1, nullptr, tbuf);
  gemm_kernel<256, 128, 0, 1, 0><<<gRows, b128, 0, stream>>>(tbuf, w2, b2, hbuf, out);
}